// EfficientEyeTracker_67791763800351
// MI455X (gfx1250) — compile-verified
//
#include <hip/hip_runtime.h>
#include <math.h>

typedef __attribute__((ext_vector_type(2))) float v2f;
typedef __attribute__((ext_vector_type(4))) float v4f;
typedef __attribute__((ext_vector_type(8))) float v8f;

#define N512 512
#define PLANE (512*512)          // 262144 floats
#define BATCH 64
#define CCH 3
#define BCPLANE (CCH*PLANE)      // 786432 floats per batch
#define CHUNK_FLOATS 8192        // per reduce block
#define CHUNKS_PER_B (BCPLANE/CHUNK_FLOATS)   // 96
#define CHUNKS_PER_PLANE (PLANE/CHUNK_FLOATS) // 32

// ---------------------------------------------------------------------------
// Kernel 1: build the 512x512 DCT-II matrix  D[k*512+n] = s(k)*cos(pi*(2n+1)k/1024)
// exact integer mod-4N argument reduction keeps cosf's argument in [0, 2*pi)
// ---------------------------------------------------------------------------
__global__ void build_dct(float* __restrict__ D) {
  int idx = blockIdx.x * blockDim.x + threadIdx.x;   // 262144 threads
  int k = idx >> 9;
  int n = idx & 511;
  int t = ((2 * n + 1) * k) & 2047;                  // mod 4N, exact
  float ang = (float)t * 3.0679615757712823e-3f;     // * pi/1024
  float s = (k == 0) ? 0.04419417382415922f          // sqrt(1/512)
                     : 0.0625f;                      // sqrt(2/512) = 1/16
  D[idx] = s * cosf(ang);
}

// ---------------------------------------------------------------------------
// Kernel 2: A_row[k,l] = att[k,l]*rw[k] ; A_col[k,l] = att[k,l]*cw[l]
// ---------------------------------------------------------------------------
__global__ void build_masks(const float* __restrict__ att,
                            const float* __restrict__ rw,
                            const float* __restrict__ cw,
                            float* __restrict__ A2) {
  int idx = blockIdx.x * blockDim.x + threadIdx.x;   // 262144 threads
  int k = idx >> 9;
  int l = idx & 511;
  float a = att[idx];
  A2[idx]         = a * rw[k];   // A_row
  A2[PLANE + idx] = a * cw[l];   // A_col
}

// ---------------------------------------------------------------------------
// Kernel 3: FP32 WMMA GEMM, C = op(A) * B, all matrices 512x512, batch of 2.
// One wave per 16x16 C tile; K-loop in steps of 4 using V_WMMA_F32_16X16X4_F32.
// ISA layouts (7.12.2): A 16x4 -> 2 VGPR, K = reg + 2*(lane>=16);
// B 4x16 -> 2 VGPR, same K rule, N = lane&15; C/D 8 VGPR, M = reg + 8*(lane>=16).
// ---------------------------------------------------------------------------
template <bool TRANSA>
__global__ void wmma_gemm512(const float* __restrict__ Abase,
                             const float* __restrict__ Bbase,
                             float* __restrict__ Cbase,
                             int strideA, int strideB, int strideC) {
  int gw   = (blockIdx.x * blockDim.x + threadIdx.x) >> 5;  // global wave id
  int lane = threadIdx.x & 31;
  int i    = gw >> 10;            // batch index (1024 tiles per matrix)
  int tile = gw & 1023;
  int tm   = (tile >> 5) << 4;    // tile row base
  int tn   = (tile & 31) << 4;    // tile col base

  const float* A = Abase + i * strideA;
  const float* B = Bbase + i * strideB;
  float*       C = Cbase + i * strideC;

  int half = lane >> 4;           // 0: lanes 0-15, 1: lanes 16-31
  int lan  = lane & 15;

  v8f acc = {};
  for (int k0 = 0; k0 < N512; k0 += 4) {
    int ka = k0 + 2 * half;       // K of VGPR0; VGPR1 holds ka+1
    v2f a, b;
    if (TRANSA) {                 // op(A)[m][k] = A[k][m]
      a.x = A[(ka + 0) * N512 + (tm + lan)];
      a.y = A[(ka + 1) * N512 + (tm + lan)];
    } else {
      a.x = A[(tm + lan) * N512 + (ka + 0)];
      a.y = A[(tm + lan) * N512 + (ka + 1)];
    }
    b.x = B[(ka + 0) * N512 + (tn + lan)];
    b.y = B[(ka + 1) * N512 + (tn + lan)];
    // (neg_a, A, neg_b, B, c_mod, C, reuse_a, reuse_b)
    acc = __builtin_amdgcn_wmma_f32_16x16x4_f32(false, a, false, b,
                                                (short)0, acc, false, false);
  }
#pragma unroll
  for (int r = 0; r < 8; ++r) {
    C[(tm + r + 8 * half) * N512 + (tn + lan)] = acc[r];
  }
}

// ---------------------------------------------------------------------------
// Kernel 4: stream x once (non-temporal: 201 MB > 192 MB L2, keep U resident);
// per block: partial dot of an 8192-float chunk of one (b,c) plane against
// U_row / U_col. Deterministic tree reduction.
// ---------------------------------------------------------------------------
__global__ void reduce_pass1(const float* __restrict__ x,
                             const float* __restrict__ Urow,
                             const float* __restrict__ Ucol,
                             float* __restrict__ partials) {
  int blk   = blockIdx.x;                 // 64*96 blocks
  int b     = blk / CHUNKS_PER_B;
  int chunk = blk % CHUNKS_PER_B;

  const v4f* X4 = (const v4f*)x;
  const v4f* R4 = (const v4f*)Urow;
  const v4f* C4 = (const v4f*)Ucol;

  long xbase4 = (long)b * (BCPLANE / 4) + (long)chunk * (CHUNK_FLOATS / 4);
  int  pbase4 = (chunk % CHUNKS_PER_PLANE) * (CHUNK_FLOATS / 4);

  int t = threadIdx.x;                    // 256 threads
  float sr = 0.0f, sc = 0.0f;
#pragma unroll
  for (int j = 0; j < 8; ++j) {
    int o = j * 256 + t;                  // 2048 float4 per chunk
    v4f xv = __builtin_nontemporal_load(&X4[xbase4 + o]);  // TH=NT stream
    v4f rv = R4[pbase4 + o];                               // hot, cached
    v4f cv = C4[pbase4 + o];                               // hot, cached
    sr += xv.x * rv.x + xv.y * rv.y + xv.z * rv.z + xv.w * rv.w;
    sc += xv.x * cv.x + xv.y * cv.y + xv.z * cv.z + xv.w * cv.w;
  }

  __shared__ float sR[256];
  __shared__ float sC[256];
  sR[t] = sr; sC[t] = sc;
  __syncthreads();
  for (int s = 128; s > 0; s >>= 1) {
    if (t < s) { sR[t] += sR[t + s]; sC[t] += sC[t + s]; }
    __syncthreads();
  }
  if (t == 0) {
    partials[blk * 2 + 0] = sC[0];
    partials[blk * 2 + 1] = sR[0];
  }
}

// ---------------------------------------------------------------------------
// Kernel 5: fixed-order final reduction + scale + sigmoid.
// out[b,0] = sigmoid(col), out[b,1] = sigmoid(row)  (stack([col,row],axis=1))
// ---------------------------------------------------------------------------
__global__ void finalize(const float* __restrict__ partials,
                         float* __restrict__ out) {
  int b = blockIdx.x * blockDim.x + threadIdx.x;
  if (b >= BATCH) return;
  float sc = 0.0f, sr = 0.0f;
  for (int j = 0; j < CHUNKS_PER_B; ++j) {
    sc += partials[(b * CHUNKS_PER_B + j) * 2 + 0];
    sr += partials[(b * CHUNKS_PER_B + j) * 2 + 1];
  }
  const float scale = 1.0f / (3.0f * 512.0f * 512.0f);
  float zc = sc * scale;
  float zr = sr * scale;
  out[2 * b + 0] = 1.0f / (1.0f + expf(-zc));
  out[2 * b + 1] = 1.0f / (1.0f + expf(-zr));
}

// ---------------------------------------------------------------------------
extern "C" void kernel_launch(void* const* d_in, const int* in_sizes, int n_in,
                              void* d_out, int out_size, void* d_ws, size_t ws_size,
                              hipStream_t stream) {
  const float* x   = (const float*)d_in[0];  // (64,3,512,512)
  const float* att = (const float*)d_in[1];  // (512,512)
  const float* rw  = (const float*)d_in[2];  // (512,)
  const float* cw  = (const float*)d_in[3];  // (512,)
  float* out = (float*)d_out;                // (64,2)

  float* ws  = (float*)d_ws;
  float* D   = ws;                     // 1 plane
  float* A2  = ws + (size_t)PLANE * 1; // 2 planes: A_row, A_col
  float* T2  = ws + (size_t)PLANE * 3; // 2 planes
  float* U2  = ws + (size_t)PLANE * 5; // 2 planes: U_row, U_col
  float* par = ws + (size_t)PLANE * 7; // 64*96*2 floats

  // 1. DCT matrix
  build_dct<<<PLANE / 256, 256, 0, stream>>>(D);
  // 2. weighted attention masks
  build_masks<<<PLANE / 256, 256, 0, stream>>>(att, rw, cw, A2);
  // 3. T[i] = D^T * A[i]   (2048 waves -> 512 blocks * 4 waves)
  wmma_gemm512<true><<<512, 128, 0, stream>>>(D, A2, T2, 0, PLANE, PLANE);
  // 4. U[i] = T[i] * D
  wmma_gemm512<false><<<512, 128, 0, stream>>>(T2, D, U2, PLANE, 0, PLANE);
  // 5. stream x (NT), partial dots vs U_row / U_col
  reduce_pass1<<<BATCH * CHUNKS_PER_B, 256, 0, stream>>>(x, U2, U2 + PLANE, par);
  // 6. final sums + sigmoid
  finalize<<<2, 32, 0, stream>>>(par, out);
}